// WeightedDNF_9156870275113
// MI455X (gfx1250) — compile-verified
//
#include <hip/hip_runtime.h>
#include <math.h>

// WeightedDNF forward, factorized for MI455X (gfx1250, wave32).
//
//   logit[b,(i,j),a,c] = nullary[b]·K0[a,c] + unary[b,i]·Ku0 + unary[b,j]·Ku1
//                      + binary[b,i,j']·Kb0 + binary[b,j,i']·Kb1 + and_bias[a,c]
//
// GEMMs run on V_WMMA_F32_16X16X4_F32 (full fp32, matches reference math).
// Sizes: B=16, O=64, C=8, A=3, K=128; P=4032; outputs 16 + 1024 + 64512 fp32.

typedef __attribute__((ext_vector_type(2))) float v2f;
typedef __attribute__((ext_vector_type(8))) float v8f;

#define THRV 0.2f
#define KDIM 128
#define NCOL 48   // 24 cols for slice-0 weights, 24 for slice-1

// One wave = one 16x16 output tile of S[M x 48].
// A is row-major [M x 128]. Weight column n (0..47): nn=n%24, a=nn/8, c=nn%8,
// k-offset base = (n<24 ? base0 : base1) into and_kernel[a][c][0..639].
//
// ISA VGPR layouts (CDNA5 ISA):
//   A 16x4 : lane<16 holds {K=k0,k0+1} for M=lane ; lane>=16 holds {k0+2,k0+3}
//   B 4x16 : lane%16 = N ; lane<16 holds {K=k0,k0+1} ; lane>=16 {k0+2,k0+3}
//   C/D    : VGPR g -> row g + 8*(lane>=16), col = lane%16
__global__ void gemm_andk_wmma(const float* __restrict__ A,
                               const float* __restrict__ andk,
                               int base0, int base1, int mTiles,
                               float* __restrict__ S)
{
    const int lane = threadIdx.x & 31;
    const int wave = blockIdx.x * (blockDim.x >> 5) + (threadIdx.x >> 5);
    if (wave >= mTiles * 3) return;                 // wave-uniform guard
    const int tm = wave / 3;
    const int tn = wave % 3;
    const int h  = lane >> 4;                       // half-wave select
    const int l  = lane & 15;

    const float* __restrict__ aPtr =
        A + (size_t)(tm * 16 + l) * KDIM + 2 * h;

    const int n  = tn * 16 + l;                     // 0..47
    const int nn = n % 24;
    const int wb = (n < 24) ? base0 : base1;
    const float* __restrict__ bPtr =
        andk + (size_t)(nn / 8) * 5120 + (size_t)(nn % 8) * 640 + wb + 2 * h;

    v8f acc = {0.f, 0.f, 0.f, 0.f, 0.f, 0.f, 0.f, 0.f};
    #pragma unroll 4
    for (int k = 0; k < KDIM; k += 4) {
        v2f av, bv;
        av.x = aPtr[k];     av.y = aPtr[k + 1];
        bv.x = bPtr[k];     bv.y = bPtr[k + 1];
        // (neg_a, A, neg_b, B, c_mod, C, reuse_a, reuse_b)
        acc = __builtin_amdgcn_wmma_f32_16x16x4_f32(
            false, av, false, bv, (short)0, acc, false, false);
    }

    float* __restrict__ o = S + (size_t)(tm * 16 + 8 * h) * NCOL + tn * 16 + l;
    #pragma unroll
    for (int g = 0; g < 8; ++g) o[(size_t)g * NCOL] = acc[g];
}

// base[b, a*8+c] = nullary[b]·andk[a,c,0:128] + thr - tanh(thr)*Σ_i andk[a,c,i]^2
__global__ void nullary_base_k(const float* __restrict__ nullary,
                               const float* __restrict__ andk,
                               float* __restrict__ baseB)
{
    const int b = blockIdx.x;       // 16
    const int t = threadIdx.x;      // 32 threads, 24 active
    if (t >= 24) return;
    const float* __restrict__ w = andk + (size_t)(t / 8) * 5120 + (size_t)(t % 8) * 640;
    float q = 0.f;
    for (int i = 0; i < 640; ++i) q += w[i] * w[i];
    float s = 0.f;
    const float* __restrict__ x = nullary + (size_t)b * KDIM;
    for (int i = 0; i < KDIM; ++i) s += x[i] * w[i];
    baseB[b * 24 + t] = s + THRV - q * tanhf(THRV);
}

// One block per (b,i). Thread j (0..62) handles permutation p=(i, jj).
__global__ void combine_k(const float* __restrict__ Sb,
                          const float* __restrict__ Su,
                          const float* __restrict__ baseB,
                          const float* __restrict__ ork,   // (3,8)
                          float* __restrict__ out,
                          float* __restrict__ pmax)
{
    const int bi = blockIdx.x;            // b*64 + i
    const int b  = bi >> 6;
    const int i  = bi & 63;
    const int j  = threadIdx.x;           // 0..63, 63 active

    __shared__ float red[63 * 16];
    __shared__ float m16[16];

    const float tb = tanhf(THRV);

    if (j < 63) {
        const int jj = j + (j >= i);              // second object id
        const int ip = (j < i) ? (i - 1) : i;     // column of (jj, i) in O-1 layout
        const float* __restrict__ r0 = Sb + ((size_t)(b * 64 + i)  * 63 + j ) * NCOL;       // Kb0 cols
        const float* __restrict__ r1 = Sb + ((size_t)(b * 64 + jj) * 63 + ip) * NCOL + 24;  // Kb1 cols
        const float* __restrict__ u0 = Su + (size_t)(b * 64 + i)  * NCOL;                   // Ku0 cols
        const float* __restrict__ u1 = Su + (size_t)(b * 64 + jj) * NCOL + 24;              // Ku1 cols
        const float* __restrict__ bb = baseB + b * 24;

        float conj[24];
        #pragma unroll
        for (int t = 0; t < 24; ++t)
            conj[t] = tanhf(bb[t] + u0[t] + u1[t] + r0[t] + r1[t]);

        // a=2 output: direct per-permutation OR stage
        float acc2 = 0.f, q2 = 0.f;
        #pragma unroll
        for (int c = 0; c < 8; ++c) {
            acc2 += conj[16 + c] * ork[16 + c];
            q2   += ork[16 + c] * ork[16 + c];
        }
        out[16 + 1024 + (size_t)bi * 63 + j] = acc2 + q2 * tb - THRV;

        // stage a=0 and a=1 slices for j-max reduction
        #pragma unroll
        for (int t = 0; t < 16; ++t) red[j * 16 + t] = conj[t];
    }
    __syncthreads();

    if (threadIdx.x < 16) {
        float m = -1e30f;
        for (int jj2 = 0; jj2 < 63; ++jj2) m = fmaxf(m, red[jj2 * 16 + threadIdx.x]);
        m16[threadIdx.x] = m;
    }
    __syncthreads();

    if (threadIdx.x < 8)                              // per-(b,i) max of a=0 slice
        pmax[(size_t)bi * 8 + threadIdx.x] = m16[threadIdx.x];

    if (threadIdx.x == 0) {                           // a=1 output for this (b,i)
        float acc1 = 0.f, q1 = 0.f;
        #pragma unroll
        for (int c = 0; c < 8; ++c) {
            acc1 += m16[8 + c] * ork[8 + c];
            q1   += ork[8 + c] * ork[8 + c];
        }
        out[16 + bi] = acc1 + q1 * tb - THRV;
    }
}

// a=0 output: reduce per-(b,i) maxima over i, then OR stage.
__global__ void final0_k(const float* __restrict__ pmax,
                         const float* __restrict__ ork,
                         float* __restrict__ out)
{
    const int b = blockIdx.x;   // 16
    if (threadIdx.x != 0) return;
    const float tb = tanhf(THRV);
    float acc = 0.f, q = 0.f;
    #pragma unroll
    for (int c = 0; c < 8; ++c) {
        float m = -1e30f;
        for (int i = 0; i < 64; ++i)
            m = fmaxf(m, pmax[((size_t)b * 64 + i) * 8 + c]);
        acc += m * ork[c];
        q   += ork[c] * ork[c];
    }
    out[b] = acc + q * tb - THRV;
}

extern "C" void kernel_launch(void* const* d_in, const int* in_sizes, int n_in,
                              void* d_out, int out_size, void* d_ws, size_t ws_size,
                              hipStream_t stream) {
    const float* nullary = (const float*)d_in[0];   // (16,128)
    const float* unary   = (const float*)d_in[1];   // (16,64,128)
    const float* binary  = (const float*)d_in[2];   // (16,64,63,128)
    const float* andk    = (const float*)d_in[3];   // (3,8,640)
    const float* ork     = (const float*)d_in[4];   // (3,8)
    float* out = (float*)d_out;                     // 16 + 1024 + 64512

    // workspace layout (fp32): Sb[64512*48] | Su[1024*48] | base[16*24] | pmax[16*64*8]
    float* ws    = (float*)d_ws;
    float* Sb    = ws;
    float* Su    = Sb + (size_t)64512 * NCOL;
    float* baseB = Su + (size_t)1024 * NCOL;
    float* pmax  = baseB + 16 * 24;

    // binary GEMM: 4032 M-tiles * 3 N-tiles = 12096 waves; 4 waves/block
    gemm_andk_wmma<<<3024, 128, 0, stream>>>(binary, andk, 384, 512, 4032, Sb);
    // unary GEMM: 64 M-tiles * 3 N-tiles = 192 waves
    gemm_andk_wmma<<<48, 128, 0, stream>>>(unary, andk, 128, 256, 64, Su);
    // nullary + and_bias base term
    nullary_base_k<<<16, 32, 0, stream>>>(nullary, andk, baseB);
    // gather + tanh + max + OR stages
    combine_k<<<1024, 64, 0, stream>>>(Sb, Su, baseB, ork, out, pmax);
    final0_k<<<16, 32, 0, stream>>>(pmax, ork, out);
}